// SequenceLayer2_75969381532484
// MI455X (gfx1250) — compile-verified
//
#include <hip/hip_runtime.h>
#include <math.h>

typedef float v2f __attribute__((ext_vector_type(2)));
typedef float v8f __attribute__((ext_vector_type(8)));

#define DM 256      // d_model
#define DS 16       // d_state
#define DR 16       // dt_rank
#define LQ 2048     // seq len
#define NB 4        // batch
#define NT (NB*LQ)  // 8192 tokens
#define CH 128      // scan chunk length
#define NCH (LQ/CH) // 16 chunks per sequence

// workspace layout (float offsets)
#define XZ_OFF  ((size_t)0)                            // 2 * NT * 512
#define XC_OFF  (XZ_OFF + (size_t)2*NT*512)            // 2 * NT * 256
#define DBC_OFF (XC_OFF + (size_t)2*NT*256)            // 2 * NT * 48
#define Y_OFF   (DBC_OFF + (size_t)2*NT*48)            // 2 * NT * 256
#define AP_OFF  (Y_OFF + (size_t)2*NT*256)             // 2*NB*NCH*DM*DS  (chunk decay)
#define HL_OFF  (AP_OFF + (size_t)2*NB*NCH*DM*DS)      // 2*NB*NCH*DM*DS  (chunk local h)
#define H0_OFF  (HL_OFF + (size_t)2*NB*NCH*DM*DS)      // 2*NB*NCH*DM*DS  (chunk init h)

__device__ __forceinline__ float sigmoidf_(float v) { return 1.0f / (1.0f + expf(-v)); }

// ---------------------------------------------------------------------------
// Kernel 1: xz[dir][t, 0:512] = x_dir @ in_w^T   (x_dir = x flipped for dir=1)
// one wave -> 16x32 tile; f32 WMMA 16x16x4, K=256
// ---------------------------------------------------------------------------
__global__ void inproj_kernel(const float* __restrict__ x,
                              const float* __restrict__ fW,
                              const float* __restrict__ rW,
                              float* __restrict__ ws) {
  const int wave = blockIdx.x * (blockDim.x >> 5) + (threadIdx.x >> 5);
  const int lane = threadIdx.x & 31;
  const int dir   = wave / (512 * 16);
  const int rem   = wave % (512 * 16);
  const int mtile = rem >> 4;
  const int npair = rem & 15;
  const int e0 = npair * 32;
  const int t0 = mtile * 16;
  const int m = lane & 15;
  const int g = lane >> 4;
  const int b  = t0 / LQ;
  const int l0 = t0 % LQ;
  int lrow = l0 + m;
  if (dir) lrow = LQ - 1 - lrow;               // flipped input for reverse dir
  const float* arow  = x + (size_t)(b * LQ + lrow) * DM;
  const float* W     = dir ? rW : fW;
  const float* b0row = W + (size_t)(e0 + m) * DM;
  const float* b1row = W + (size_t)(e0 + 16 + m) * DM;
  v8f c0 = {}; v8f c1 = {};
  for (int kk = 0; kk < DM; kk += 4) {
    const int kb = kk + 2 * g;                 // lane's 2 elems contiguous in k
    v2f a  = *(const v2f*)(arow  + kb);
    v2f b0 = *(const v2f*)(b0row + kb);
    v2f b1 = *(const v2f*)(b1row + kb);
    c0 = __builtin_amdgcn_wmma_f32_16x16x4_f32(false, a, false, b0, (short)0, c0, false, false);
    c1 = __builtin_amdgcn_wmma_f32_16x16x4_f32(false, a, false, b1, (short)0, c1, false, false);
  }
  float* xz = ws + XZ_OFF + (size_t)dir * NT * 512;
#pragma unroll
  for (int v = 0; v < 8; ++v) {
    const int row = t0 + v + 8 * g;
    xz[(size_t)row * 512 + e0 + m]      = c0[v];
    xz[(size_t)row * 512 + e0 + 16 + m] = c1[v];
  }
}

// ---------------------------------------------------------------------------
// Kernel 2: depthwise conv (D_CONV=2) + SiLU -> xc buffer
// ---------------------------------------------------------------------------
__global__ void conv_silu_kernel(const float* __restrict__ fcw, const float* __restrict__ fcb,
                                 const float* __restrict__ rcw, const float* __restrict__ rcb,
                                 float* __restrict__ ws) {
  const size_t idx = (size_t)blockIdx.x * blockDim.x + threadIdx.x; // 2*NT*DM threads
  const int dir = (int)(idx / ((size_t)NT * DM));
  const size_t rem = idx % ((size_t)NT * DM);
  const int t = (int)(rem / DM);
  const int d = (int)(rem % DM);
  const int l = t % LQ;
  const float* xz = ws + XZ_OFF + (size_t)dir * NT * 512;
  const float cur  = xz[(size_t)t * 512 + d];
  const float prev = (l == 0) ? 0.0f : xz[(size_t)(t - 1) * 512 + d];
  const float* cw = dir ? rcw : fcw;
  const float* cb = dir ? rcb : fcb;
  const float v = prev * cw[2 * d] + cur * cw[2 * d + 1] + cb[d];
  float* xc = ws + XC_OFF + (size_t)dir * NT * DM;
  xc[(size_t)t * DM + d] = v * sigmoidf_(v);
}

// ---------------------------------------------------------------------------
// Kernel 3: dbc[dir][t, 0:48] = xc @ xproj_w^T  (WMMA, N = 3 tiles of 16)
// ---------------------------------------------------------------------------
__global__ void xproj_kernel(const float* __restrict__ fXW,
                             const float* __restrict__ rXW,
                             float* __restrict__ ws) {
  const int wave = blockIdx.x * (blockDim.x >> 5) + (threadIdx.x >> 5);
  const int lane = threadIdx.x & 31;
  const int dir   = wave / (512 * 3);
  const int rem   = wave % (512 * 3);
  const int mtile = rem / 3;
  const int ntile = rem % 3;
  const int n0 = ntile * 16;
  const int t0 = mtile * 16;
  const int m = lane & 15;
  const int g = lane >> 4;
  const float* xc = ws + XC_OFF + (size_t)dir * NT * DM;
  const float* W  = dir ? rXW : fXW;
  const float* arow = xc + (size_t)(t0 + m) * DM;
  const float* brow = W + (size_t)(n0 + m) * DM;
  v8f c = {};
  for (int kk = 0; kk < DM; kk += 4) {
    const int kb = kk + 2 * g;
    v2f a  = *(const v2f*)(arow + kb);
    v2f bb = *(const v2f*)(brow + kb);
    c = __builtin_amdgcn_wmma_f32_16x16x4_f32(false, a, false, bb, (short)0, c, false, false);
  }
  float* dbc = ws + DBC_OFF + (size_t)dir * NT * 48;
#pragma unroll
  for (int v = 0; v < 8; ++v) {
    const int row = t0 + v + 8 * g;
    dbc[(size_t)row * 48 + n0 + m] = c[v];
  }
}

// ---------------------------------------------------------------------------
// Kernel 4a (scan pass A): per-chunk local recurrence with h0 = 0.
// Emits per (dir,b,chunk,d): chunk decay Aprod[n] = exp(A_n * sum(dt)) and
// local final state h_loc[n]. block = 256 channels of one (dir,b,chunk).
// ---------------------------------------------------------------------------
__global__ void scanA_kernel(const float* __restrict__ fdtw, const float* __restrict__ fdtb,
                             const float* __restrict__ fAlog,
                             const float* __restrict__ rdtw, const float* __restrict__ rdtb,
                             const float* __restrict__ rAlog,
                             float* __restrict__ ws) {
  __shared__ float sdbc[CH * 48];                       // 24 KB
  const int blk = blockIdx.x;                           // 2*NB*NCH = 128
  const int dir = blk / (NB * NCH);
  const int rem = blk % (NB * NCH);
  const int b   = rem / NCH;
  const int c   = rem % NCH;
  const int d   = threadIdx.x;                          // 256

  const float* dtw  = (dir ? rdtw  : fdtw)  + (size_t)d * DR;
  const float  dtb  = (dir ? rdtb  : fdtb)[d];
  const float* Alog = (dir ? rAlog : fAlog) + (size_t)d * DS;

  float w[DR], An[DS], h[DS];
#pragma unroll
  for (int r = 0; r < DR; ++r) w[r] = dtw[r];
#pragma unroll
  for (int n = 0; n < DS; ++n) { An[n] = -expf(Alog[n]); h[n] = 0.0f; }

  const float* xc  = ws + XC_OFF  + (size_t)dir * NT * DM;
  const float* dbc = ws + DBC_OFF + (size_t)dir * NT * 48
                       + (size_t)(b * LQ + c * CH) * 48;
  for (int i = threadIdx.x; i < CH * 48; i += 256) sdbc[i] = dbc[i];
  __syncthreads();

  float dtsum = 0.0f;
  for (int li = 0; li < CH; ++li) {
    const int t = b * LQ + c * CH + li;
    const float xcv = xc[(size_t)t * DM + d];
    const float* row = &sdbc[li * 48];
    float u = dtb;
#pragma unroll
    for (int r = 0; r < DR; ++r) u += row[r] * w[r];
    const float dt = (u > 20.0f) ? u : log1pf(expf(u));  // softplus
    dtsum += dt;
#pragma unroll
    for (int n = 0; n < DS; ++n) {
      const float dA = expf(dt * An[n]);
      h[n] = dA * h[n] + (dt * row[16 + n]) * xcv;
    }
  }
  const size_t base = ((size_t)(dir * NB + b) * NCH + c) * DM * DS + (size_t)d * DS;
  float* Ap = ws + AP_OFF;
  float* Hl = ws + HL_OFF;
#pragma unroll
  for (int n = 0; n < DS; ++n) {
    Ap[base + n] = expf(dtsum * An[n]);   // prod_t exp(dt_t*A_n) = exp(A_n*sum dt)
    Hl[base + n] = h[n];
  }
}

// ---------------------------------------------------------------------------
// Kernel 4b (scan pass B): sequential combine over the 16 chunks.
// one thread per (dir,b,d,n): h0[c] = Aprod[c-1]*h0[c-1] + hloc[c-1], h0[0]=0
// ---------------------------------------------------------------------------
__global__ void scanB_kernel(float* __restrict__ ws) {
  const int tid = blockIdx.x * blockDim.x + threadIdx.x;  // 2*NB*DM*DS = 32768
  const int dn  = tid & (DM * DS - 1);                    // d*16+n (coalesced)
  const int sb  = tid >> 12;                              // dir*NB + b
  const float* Ap = ws + AP_OFF;
  const float* Hl = ws + HL_OFF;
  float*       H0 = ws + H0_OFF;
  float h = 0.0f;
  for (int c = 0; c < NCH; ++c) {
    const size_t idx = ((size_t)(sb * NCH + c)) * DM * DS + dn;
    H0[idx] = h;
    h = Ap[idx] * h + Hl[idx];
  }
}

// ---------------------------------------------------------------------------
// Kernel 4c (scan pass C): full within-chunk recurrence from correct h0,
// fused with D-skip and SiLU(z) gating -> y buffer.
// ---------------------------------------------------------------------------
__global__ void scanC_kernel(const float* __restrict__ fdtw, const float* __restrict__ fdtb,
                             const float* __restrict__ fAlog, const float* __restrict__ fDp,
                             const float* __restrict__ rdtw, const float* __restrict__ rdtb,
                             const float* __restrict__ rAlog, const float* __restrict__ rDp,
                             float* __restrict__ ws) {
  __shared__ float sdbc[CH * 48];                        // 24 KB
  const int blk = blockIdx.x;                            // 2*NB*NCH = 128
  const int dir = blk / (NB * NCH);
  const int rem = blk % (NB * NCH);
  const int b   = rem / NCH;
  const int c   = rem % NCH;
  const int d   = threadIdx.x;                           // 256

  const float* dtw  = (dir ? rdtw  : fdtw)  + (size_t)d * DR;
  const float  dtb  = (dir ? rdtb  : fdtb)[d];
  const float* Alog = (dir ? rAlog : fAlog) + (size_t)d * DS;
  const float  Dpv  = (dir ? rDp   : fDp)[d];

  const size_t base = ((size_t)(dir * NB + b) * NCH + c) * DM * DS + (size_t)d * DS;
  const float* H0 = ws + H0_OFF;

  float w[DR], An[DS], h[DS];
#pragma unroll
  for (int r = 0; r < DR; ++r) w[r] = dtw[r];
#pragma unroll
  for (int n = 0; n < DS; ++n) { An[n] = -expf(Alog[n]); h[n] = H0[base + n]; }

  const float* xc  = ws + XC_OFF  + (size_t)dir * NT * DM;
  const float* xz  = ws + XZ_OFF  + (size_t)dir * NT * 512;
  const float* dbc = ws + DBC_OFF + (size_t)dir * NT * 48
                       + (size_t)(b * LQ + c * CH) * 48;
  float*       y   = ws + Y_OFF   + (size_t)dir * NT * DM;

  for (int i = threadIdx.x; i < CH * 48; i += 256) sdbc[i] = dbc[i];
  __syncthreads();

  for (int li = 0; li < CH; ++li) {
    const int t = b * LQ + c * CH + li;
    const float xcv = xc[(size_t)t * DM + d];
    const float* row = &sdbc[li * 48];
    float u = dtb;
#pragma unroll
    for (int r = 0; r < DR; ++r) u += row[r] * w[r];
    const float dt = (u > 20.0f) ? u : log1pf(expf(u));  // softplus
    float ysum = 0.0f;
#pragma unroll
    for (int n = 0; n < DS; ++n) {
      const float dA = expf(dt * An[n]);
      h[n] = dA * h[n] + (dt * row[16 + n]) * xcv;       // h = dA*h + dt*B*x
      ysum += h[n] * row[32 + n];                        // y += h*C
    }
    const float zv = xz[(size_t)t * 512 + 256 + d];
    y[(size_t)t * DM + d] = (ysum + Dpv * xcv) * (zv * sigmoidf_(zv));
  }
}

// ---------------------------------------------------------------------------
// Kernel 5: out[t,:] = y_f[t] @ f_out_w^T + y_r[flip(t)] @ r_out_w^T
// ---------------------------------------------------------------------------
__global__ void outproj_kernel(const float* __restrict__ fOW,
                               const float* __restrict__ rOW,
                               const float* __restrict__ ws,
                               float* __restrict__ out) {
  const int wave = blockIdx.x * (blockDim.x >> 5) + (threadIdx.x >> 5);
  const int lane = threadIdx.x & 31;
  const int mtile = wave >> 3;
  const int npair = wave & 7;
  const int e0 = npair * 32;
  const int t0 = mtile * 16;
  const int m = lane & 15;
  const int g = lane >> 4;
  const int b  = t0 / LQ;
  const int l0 = t0 % LQ;

  const float* y0 = ws + Y_OFF;                       // forward, natural order
  const float* y1 = ws + Y_OFF + (size_t)NT * DM;     // reverse, flipped order

  v8f c0 = {}; v8f c1 = {};

  // forward contribution
  {
    const float* arow  = y0 + (size_t)(t0 + m) * DM;
    const float* b0row = fOW + (size_t)(e0 + m) * DM;
    const float* b1row = fOW + (size_t)(e0 + 16 + m) * DM;
    for (int kk = 0; kk < DM; kk += 4) {
      const int kb = kk + 2 * g;
      v2f a  = *(const v2f*)(arow  + kb);
      v2f b0 = *(const v2f*)(b0row + kb);
      v2f b1 = *(const v2f*)(b1row + kb);
      c0 = __builtin_amdgcn_wmma_f32_16x16x4_f32(false, a, false, b0, (short)0, c0, false, false);
      c1 = __builtin_amdgcn_wmma_f32_16x16x4_f32(false, a, false, b1, (short)0, c1, false, false);
    }
  }
  // reverse contribution (read y1 at flipped sequence position)
  {
    const int lf = LQ - 1 - (l0 + m);
    const float* arow  = y1 + (size_t)(b * LQ + lf) * DM;
    const float* b0row = rOW + (size_t)(e0 + m) * DM;
    const float* b1row = rOW + (size_t)(e0 + 16 + m) * DM;
    for (int kk = 0; kk < DM; kk += 4) {
      const int kb = kk + 2 * g;
      v2f a  = *(const v2f*)(arow  + kb);
      v2f b0 = *(const v2f*)(b0row + kb);
      v2f b1 = *(const v2f*)(b1row + kb);
      c0 = __builtin_amdgcn_wmma_f32_16x16x4_f32(false, a, false, b0, (short)0, c0, false, false);
      c1 = __builtin_amdgcn_wmma_f32_16x16x4_f32(false, a, false, b1, (short)0, c1, false, false);
    }
  }
#pragma unroll
  for (int v = 0; v < 8; ++v) {
    const int row = t0 + v + 8 * g;
    out[(size_t)row * DM + e0 + m]      = c0[v];
    out[(size_t)row * DM + e0 + 16 + m] = c1[v];
  }
}

// ---------------------------------------------------------------------------
extern "C" void kernel_launch(void* const* d_in, const int* in_sizes, int n_in,
                              void* d_out, int out_size, void* d_ws, size_t ws_size,
                              hipStream_t stream) {
  const float* x         = (const float*)d_in[0];
  const float* f_in_w    = (const float*)d_in[1];
  const float* f_conv_w  = (const float*)d_in[2];
  const float* f_conv_b  = (const float*)d_in[3];
  const float* f_xproj_w = (const float*)d_in[4];
  const float* f_dt_w    = (const float*)d_in[5];
  const float* f_dt_b    = (const float*)d_in[6];
  const float* f_A_log   = (const float*)d_in[7];
  const float* f_D       = (const float*)d_in[8];
  const float* f_out_w   = (const float*)d_in[9];
  const float* r_in_w    = (const float*)d_in[10];
  const float* r_conv_w  = (const float*)d_in[11];
  const float* r_conv_b  = (const float*)d_in[12];
  const float* r_xproj_w = (const float*)d_in[13];
  const float* r_dt_w    = (const float*)d_in[14];
  const float* r_dt_b    = (const float*)d_in[15];
  const float* r_A_log   = (const float*)d_in[16];
  const float* r_D       = (const float*)d_in[17];
  const float* r_out_w   = (const float*)d_in[18];

  float* ws  = (float*)d_ws;
  float* out = (float*)d_out;

  // 1) in-projection: 2*512*16 = 16384 waves, 8 waves/block
  inproj_kernel<<<2048, 256, 0, stream>>>(x, f_in_w, r_in_w, ws);
  // 2) conv + SiLU: 2*NT*DM threads
  conv_silu_kernel<<<16384, 256, 0, stream>>>(f_conv_w, f_conv_b, r_conv_w, r_conv_b, ws);
  // 3) x-projection: 2*512*3 = 3072 waves
  xproj_kernel<<<384, 256, 0, stream>>>(f_xproj_w, r_xproj_w, ws);
  // 4) chunked selective scan: A (chunk summaries), B (combine), C (final)
  scanA_kernel<<<2 * NB * NCH, 256, 0, stream>>>(f_dt_w, f_dt_b, f_A_log,
                                                 r_dt_w, r_dt_b, r_A_log, ws);
  scanB_kernel<<<(2 * NB * DM * DS) / 256, 256, 0, stream>>>(ws);
  scanC_kernel<<<2 * NB * NCH, 256, 0, stream>>>(f_dt_w, f_dt_b, f_A_log, f_D,
                                                 r_dt_w, r_dt_b, r_A_log, r_D, ws);
  // 5) out-projection + bidirectional merge: 512*8 = 4096 waves
  outproj_kernel<<<512, 256, 0, stream>>>(f_out_w, r_out_w, ws, out);
}